// BarycentricCoordinates_36464272343399
// MI455X (gfx1250) — compile-verified
//
#include <hip/hip_runtime.h>

// Problem constants from the reference: R=5, A=8, N=16, M=15.
#define RR 5
#define AA 8
#define NN 16
#define MM 15

typedef float v2f __attribute__((ext_vector_type(2)));
typedef float v8f __attribute__((ext_vector_type(8)));

static __device__ __forceinline__ float perm_f32(int dstLane, float v) {
    int r = __builtin_amdgcn_ds_permute(dstLane << 2, __builtin_bit_cast(int, v));
    return __builtin_bit_cast(float, r);
}

__global__ __launch_bounds__(256) void bc_wmma_kernel(
    const float* __restrict__ tpl,   // (R, A, 2)
    const float* __restrict__ proj,  // (V, N, 2)
    float* __restrict__ wout,        // (V, R, A, 3) f32
    int* __restrict__ iout,          // (V, R, A, 3) i32
    int nTasks)
{
    // Per-wave 16x16 Gram matrix scratch, stored column-major:
    // G_lds[col*16 + row] == G[row][col]
    __shared__ float Gl[8 * 256];

    const int tid   = threadIdx.x;
    const int lane  = tid & 31;
    const int wv    = tid >> 5;       // wave index within block (0..7)
    const int j     = lane & 15;      // point index (upper half mirrors lower)
    const int upper = lane & 16;

    int task = blockIdx.x * 8 + wv;
    const bool valid = (task < nTasks);
    if (task >= nTasks) task = nTasks - 1;   // wave-uniform clamp: EXEC untouched

    const int v  = task / (RR * AA);
    const int ra = task - v * (RR * AA);

    // Template point (wave-uniform -> scalar loads)
    const float tx = tpl[ra * 2 + 0];
    const float ty = tpl[ra * 2 + 1];

    // Projection point j for view v
    const float px = proj[(v * NN + j) * 2 + 0];
    const float py = proj[(v * NN + j) * 2 + 1];

    // Squared distance (monotonic with the reference's norm)
    const float dx = px - tx, dy = py - ty;
    const float d2 = dx * dx + dy * dy;

    // Stable rank of point j among the 16 (ties -> lower original index first)
    int rank = 0;
#pragma unroll
    for (int i = 0; i < NN; ++i) {
        const float di = __shfl(d2, i, 32);
        rank += (di < d2 || (di == d2 && i < j)) ? 1 : 0;
    }

    // Scatter points into sorted order via ds_permute.
    // Upper half-wave scatters into lanes 16..31 (mirror) to avoid collisions.
    const int dst = rank + upper;
    const float sx   = perm_f32(dst, px);
    const float sy   = perm_f32(dst, py);
    const int   sidx = __builtin_amdgcn_ds_permute(dst << 2, j);

    // Closest projection (rank 0)
    const float cx = __shfl(sx, 0, 32);
    const float cy = __shfl(sy, 0, 32);

    // Row vector for the Gram WMMA:
    //   rows 0..14 : v0[r] = sorted[r+1] - closest
    //   row  15    : v2    = template - closest
    const float nxs = __shfl(sx, j + 1, 32);
    const float nys = __shfl(sy, j + 1, 32);
    const float rx = (j == MM) ? (tx - cx) : (nxs - cx);
    const float ry = (j == MM) ? (ty - cy) : (nys - cy);

    // A (16x4 f32): lanes 0-15 carry K=0,1; lanes 16-31 carry K=2,3 (zero pad).
    // B (4x16 f32): lanes 0-15 carry rows K=0,1; lanes 16-31 rows K=2,3 (zero).
    // G is symmetric, so the same register pair serves as both A and B.
    v2f ab;
    ab.x = upper ? 0.0f : rx;
    ab.y = upper ? 0.0f : ry;
    v8f acc = {};
    // G = X * X^T : one 16x16x4 f32 WMMA computes dot00 (diag), dot01, dot02 (row/col 15)
    acc = __builtin_amdgcn_wmma_f32_16x16x4_f32(
        /*neg_a=*/false, ab, /*neg_b=*/false, ab,
        /*c_mod=*/(short)0, acc, /*reuse_a=*/false, /*reuse_b=*/false);

    // C/D layout: lane -> column (lane&15), VGPR q -> row q (+8 for upper half).
    // Store column-major so each lane writes 8 consecutive floats.
    const int rowBase = upper ? 8 : 0;
    float* gcol = &Gl[wv * 256 + j * 16 + rowBase];
#pragma unroll
    for (int q = 0; q < 8; ++q) gcol[q] = acc[q];

    __syncthreads();

    const float* gw = &Gl[wv * 256];
    const float INFV = __builtin_inff();

    // Evaluate the 225 (n,m) candidate pairs, strided across the 32 lanes.
    float bs = INFV;
    int   bf = 0x7fffffff;
    for (int k = lane; k < MM * MM; k += 32) {
        const int n = k / MM;
        const int m = k - n * MM;
        const float d00n = gw[n * 16 + n];
        const float d00m = gw[m * 16 + m];
        const float d01  = gw[m * 16 + n];        // G[n][m]
        const float d02n = gw[15 * 16 + n];       // G[n][15]
        const float d02m = gw[15 * 16 + m];
        float den = d00n * d00m - d01 * d01;
        den = (den == 0.0f) ? 1e-10f : den;
        const float p2 = (d02n * d00m - d01 * d02m) / den;
        const float p1 = (d00n * d02m - d01 * d02n) / den;
        const float p0 = 1.0f - p2 - p1;
        const float w0 = (p0 <= 0.0f) ? INFV : p0;
        const float w1 = (p2 <= 0.0f) ? INFV : p2;
        const float w2 = (p1 <= 0.0f) ? INFV : p1;
        const float s = fmaxf(fmaxf(w0 * w0, w1 * w1), w2 * w2);
        if (s < bs || (s == bs && k < bf)) { bs = s; bf = k; }
    }

    // Wave-wide argmin with first-index tie-break (matches jnp.argmin)
#pragma unroll
    for (int msk = 16; msk >= 1; msk >>= 1) {
        const float s2 = __shfl_xor(bs, msk, 32);
        const int   f2 = __shfl_xor(bf, msk, 32);
        if (s2 < bs || (s2 == bs && f2 < bf)) { bs = s2; bf = f2; }
    }

    // Recompute winning weights (bf is wave-uniform now)
    const int n = bf / MM;
    const int m = bf - n * MM;
    {
        const float d00n = gw[n * 16 + n];
        const float d00m = gw[m * 16 + m];
        const float d01  = gw[m * 16 + n];
        const float d02n = gw[15 * 16 + n];
        const float d02m = gw[15 * 16 + m];
        float den = d00n * d00m - d01 * d01;
        den = (den == 0.0f) ? 1e-10f : den;
        const float p2 = (d02n * d00m - d01 * d02m) / den;
        const float p1 = (d00n * d02m - d01 * d02n) / den;
        const float p0 = 1.0f - p2 - p1;
        float w0 = (p0 <= 0.0f) ? INFV : p0;
        float w1 = (p2 <= 0.0f) ? INFV : p2;
        float w2 = (p1 <= 0.0f) ? INFV : p1;
        const bool bad = (w0 == INFV) || (w1 == INFV) || (w2 == INFV);
        if (bad) { w0 = 0.0f; w1 = 0.0f; w2 = 0.0f; }

        // Original projection indices: rank 0 (closest), rank n+1, rank m+1
        const int i0 = __shfl(sidx, 0, 32);
        const int i1 = __shfl(sidx, n + 1, 32);
        const int i2 = __shfl(sidx, m + 1, 32);

        if (valid && lane == 0) {
            wout[task * 3 + 0] = w0;   // p0
            wout[task * 3 + 1] = w1;   // p2
            wout[task * 3 + 2] = w2;   // p1
            iout[task * 3 + 0] = i0;
            iout[task * 3 + 1] = i1;
            iout[task * 3 + 2] = i2;
        }
    }
}

extern "C" void kernel_launch(void* const* d_in, const int* in_sizes, int n_in,
                              void* d_out, int out_size, void* d_ws, size_t ws_size,
                              hipStream_t stream) {
    (void)n_in; (void)d_ws; (void)ws_size; (void)out_size;
    const float* tpl  = (const float*)d_in[0];   // (R, A, 2) f32
    const float* proj = (const float*)d_in[1];   // (V, N, 2) f32

    const int V = in_sizes[1] / (NN * 2);
    const int nTasks = V * RR * AA;

    float* wout = (float*)d_out;                           // first output: weights
    int*   iout = (int*)((float*)d_out + (size_t)nTasks * 3); // second output: indices

    const int wavesPerBlock = 8;                  // 256 threads = 8 wave32
    const int blocks = (nTasks + wavesPerBlock - 1) / wavesPerBlock;
    bc_wmma_kernel<<<blocks, 256, 0, stream>>>(tpl, proj, wout, iout, nTasks);
}